// PredictiveVQTSS_2259152797974
// MI455X (gfx1250) — compile-verified
//
#include <hip/hip_runtime.h>
#include <hip/hip_bf16.h>

#define Tlen 8192
#define Hdim 256
#define Kcb  512
#define Bsz  16

#define CONV_BLK_N 64
#define XS_W 96   // 64 + 2*max_halo, max_halo = ((5-1)/2)*8 = 16

typedef __attribute__((ext_vector_type(2))) float v2f;
typedef __attribute__((ext_vector_type(8))) float v8f;

#if defined(__HIP_DEVICE_COMPILE__) && \
    __has_builtin(__builtin_amdgcn_global_load_async_to_lds_b32) && \
    __has_builtin(__builtin_amdgcn_s_wait_asynccnt)
#define USE_ASYNC_LDS 1
#else
#define USE_ASYNC_LDS 0
#endif

// ---------------------------------------------------------------------------
// Fold conv-bias + eval BatchNorm into per-channel scale/shift:
//   y = (conv + b - m) * g/sqrt(v+eps) + beta  =  conv*sc + sh
// ---------------------------------------------------------------------------
__global__ void bn_fold_kernel(const float* __restrict__ cb,
                               const float* __restrict__ g,
                               const float* __restrict__ beta,
                               const float* __restrict__ mean,
                               const float* __restrict__ var,
                               float* __restrict__ sc, float* __restrict__ sh) {
  int i = threadIdx.x;  // 256 threads == Hdim
  float s = g[i] * rsqrtf(var[i] + 1e-5f);
  sc[i] = s;
  sh[i] = (cb[i] - mean[i]) * s + beta[i];
}

// Codebook squared norms: e2[k] = sum_h emb[k,h]^2
__global__ void code_norm_kernel(const float* __restrict__ emb, float* __restrict__ e2) {
  int k = blockIdx.x * blockDim.x + threadIdx.x;
  if (k < Kcb) {
    float s = 0.f;
    for (int h = 0; h < Hdim; ++h) { float v = emb[(size_t)k * Hdim + h]; s += v * v; }
    e2[k] = s;
  }
}

// Weight transpose: w[O=256][C][k] -> wt[k][C][O]  (coalesced A-matrix loads)
__global__ void wtrans_kernel(const float* __restrict__ w, float* __restrict__ wt,
                              int C, int k) {
  int gid = blockIdx.x * 256 + threadIdx.x;
  int total = 256 * C * k;
  if (gid >= total) return;
  int o = gid % 256;
  int rest = gid / 256;
  int ci = rest % C;
  int j = rest / C;
  wt[gid] = w[((size_t)o * C + ci) * k + j];
}

// Codebook transpose: emb[K][H] -> embT[H][K]  (coalesced B-matrix loads)
__global__ void etrans_kernel(const float* __restrict__ emb, float* __restrict__ embT) {
  int gid = blockIdx.x * 256 + threadIdx.x;  // over H*K
  int k = gid % Kcb;
  int h = gid / Kcb;
  embT[gid] = emb[(size_t)k * Hdim + h];
}

// ---------------------------------------------------------------------------
// Dilated conv1d as per-tap GEMM via V_WMMA_F32_16X16X4_F32.
// Block: 256 threads = 8 waves; block tile = 128 out-ch x 64 time.
// Wave tile: 16 out-ch x 64 time (4 N-sub-tiles reuse one A register pair).
// Activation slab for each 4-channel K-chunk (incl. dilation halo) is staged
// in double-buffered LDS -- via GLOBAL_LOAD_ASYNC_TO_LDS when available --
// so the next chunk's copy overlaps the current chunk's WMMA stream.
// WMMA f32 16x16x4 layouts (ISA 7.12.2):
//   A 16x4 : lanes 0-15 M=0..15 hold K={0,1}; lanes 16-31 same M, K={2,3}
//   B 4x16 : lanes 0-15 N=0..15 hold K={0,1}; lanes 16-31 same N, K={2,3}
//   C/D    : vgpr r -> lanes 0-15 (M=r, N=lane); lanes 16-31 (M=8+r, N=lane-16)
// ---------------------------------------------------------------------------
__global__ __launch_bounds__(256)
void conv_wmma_kernel(const float* __restrict__ x,      // [B, C_in, T]
                      const float* __restrict__ wt,     // [ktaps][C_in][256]
                      const float* __restrict__ scale,  // [256] or null (==1)
                      const float* __restrict__ shift,  // [256]
                      const float* __restrict__ resid,  // [B, 256, T] or null
                      float* __restrict__ out,          // [B, 256, T]
                      int C_in, int ktaps, int dil, int do_relu) {
  __shared__ float xs[2 * 4 * XS_W];   // double buffer

  const int tid  = threadIdx.x;
  const int lane = tid & 31;
  const int wave = tid >> 5;
  const int half = lane >> 4;
  const int l15  = lane & 15;
  const int m0 = blockIdx.y * 128 + wave * 16;
  const int n0 = blockIdx.x * CONV_BLK_N;
  const int b  = blockIdx.z;
  const int center = (ktaps - 1) >> 1;
  const int halo   = center * dil;
  const int width  = CONV_BLK_N + 2 * halo;   // <= 96
  const int nch    = C_in >> 2;               // 4-channel K-chunks

  v8f acc0 = {0.f,0.f,0.f,0.f,0.f,0.f,0.f,0.f};
  v8f acc1 = acc0, acc2 = acc0, acc3 = acc0;

  const int ck = half * 2;  // K-slice base for this lane half

  auto stage = [&](int chunk, int bufsel) {
    float* dst = xs + bufsel * (4 * XS_W);
    const int c = chunk * 4;
    for (int i = tid; i < 4 * width; i += 256) {
      const int ci = i / width;
      const int tt = i - ci * width;
      const int t  = n0 - halo + tt;
      const bool inb = (t >= 0) && (t < Tlen);
#if USE_ASYNC_LDS
      if (inb) {
        __builtin_amdgcn_global_load_async_to_lds_b32(
            (__attribute__((address_space(1))) int*)(x + ((size_t)b * C_in + c + ci) * Tlen + t),
            (__attribute__((address_space(3))) int*)(dst + ci * XS_W + tt),
            0, 0);
      } else {
        dst[ci * XS_W + tt] = 0.f;
      }
#else
      dst[ci * XS_W + tt] =
          inb ? x[((size_t)b * C_in + c + ci) * Tlen + t] : 0.f;
#endif
    }
  };

  stage(0, 0);
  for (int chunk = 0; chunk < nch; ++chunk) {
#if USE_ASYNC_LDS
    __builtin_amdgcn_s_wait_asynccnt(0);   // drain this wave's async copies
#endif
    __syncthreads();                       // chunk staged; prior buffer free
    if (chunk + 1 < nch) stage(chunk + 1, (chunk + 1) & 1);

    const float* buf = xs + (chunk & 1) * (4 * XS_W);
    const int c = chunk * 4;
    for (int j = 0; j < ktaps; ++j) {
      v2f a;
      const float* wr = wt + ((size_t)j * C_in + (c + ck)) * Hdim + m0 + l15;
      a.x = wr[0];
      a.y = wr[Hdim];
      const float* xr0 = buf + (ck + 0) * XS_W + j * dil + l15;
      const float* xr1 = buf + (ck + 1) * XS_W + j * dil + l15;
      v2f bb;
      bb.x = xr0[0];  bb.y = xr1[0];
      acc0 = __builtin_amdgcn_wmma_f32_16x16x4_f32(false, a, false, bb, (short)0, acc0, false, false);
      bb.x = xr0[16]; bb.y = xr1[16];
      acc1 = __builtin_amdgcn_wmma_f32_16x16x4_f32(false, a, false, bb, (short)0, acc1, false, false);
      bb.x = xr0[32]; bb.y = xr1[32];
      acc2 = __builtin_amdgcn_wmma_f32_16x16x4_f32(false, a, false, bb, (short)0, acc2, false, false);
      bb.x = xr0[48]; bb.y = xr1[48];
      acc3 = __builtin_amdgcn_wmma_f32_16x16x4_f32(false, a, false, bb, (short)0, acc3, false, false);
    }
  }

#pragma unroll
  for (int r = 0; r < 8; ++r) {
    const int mch = m0 + half * 8 + r;
    const float scv = scale ? scale[mch] : 1.f;
    const float shv = shift[mch];
    const size_t obase = ((size_t)b * Hdim + mch) * Tlen + n0 + l15;
    float v;

    v = acc0[r] * scv + shv;
    if (resid) v += resid[obase + 0];
    if (do_relu) v = fmaxf(v, 0.f);
    out[obase + 0] = v;

    v = acc1[r] * scv + shv;
    if (resid) v += resid[obase + 16];
    if (do_relu) v = fmaxf(v, 0.f);
    out[obase + 16] = v;

    v = acc2[r] * scv + shv;
    if (resid) v += resid[obase + 32];
    if (do_relu) v = fmaxf(v, 0.f);
    out[obase + 32] = v;

    v = acc3[r] * scv + shv;
    if (resid) v += resid[obase + 48];
    if (do_relu) v = fmaxf(v, 0.f);
    out[obase + 48] = v;
  }
}

// ---------------------------------------------------------------------------
// VQ argmin. One block per 16-row group ((b,t) rows, contiguous t).
// The 16x256 activation slice is staged once in LDS; the 8 waves each sweep
// 64 codes (4 WMMA tiles) with coalesced reads of the transposed codebook.
// d = |e|^2 - 2 x.e  (|x|^2 constant per row -> irrelevant for argmin).
// Reduction: butterfly min within 16-lane halves, then cross-wave LDS reduce
// in ascending wave order (preserves lowest-index tie-break).
// ---------------------------------------------------------------------------
__global__ __launch_bounds__(256)
void vq_argmin_kernel(const float* __restrict__ z,     // [B, H, T]
                      const float* __restrict__ embT,  // [H, K]
                      const float* __restrict__ e2,    // [K]
                      int* __restrict__ idx) {         // [B*T]
  __shared__ float zs[Hdim * 16];   // zs[ch*16 + row], 16 KB
  __shared__ float redv[8 * 16];
  __shared__ int   redi[8 * 16];

  const int tid  = threadIdx.x;
  const int lane = tid & 31;
  const int wave = tid >> 5;
  const int half = lane >> 4;
  const int l15  = lane & 15;
  const int row0 = blockIdx.x * 16;
  const int b  = row0 / Tlen;
  const int t0 = row0 % Tlen;  // T % 16 == 0: group never crosses batches

  for (int i = tid; i < Hdim * 16; i += 256) {
    const int ch = i >> 4, row = i & 15;
    zs[i] = z[((size_t)b * Hdim + ch) * Tlen + t0 + row];
  }
  __syncthreads();

  float minv[8];
  int   mini[8];
#pragma unroll
  for (int r = 0; r < 8; ++r) { minv[r] = 3.402823466e38f; mini[r] = 0; }

  const int ck = half * 2;
  const int nbase = wave * 64;
  for (int nt = 0; nt < 4; ++nt) {
    const int n0 = nbase + nt * 16;
    v8f acc = {0.f,0.f,0.f,0.f,0.f,0.f,0.f,0.f};
    for (int c = 0; c < Hdim; c += 4) {
      v2f a, bb;
      a.x = zs[(c + ck + 0) * 16 + l15];
      a.y = zs[(c + ck + 1) * 16 + l15];
      bb.x = embT[(size_t)(c + ck + 0) * Kcb + n0 + l15];
      bb.y = embT[(size_t)(c + ck + 1) * Kcb + n0 + l15];
      acc = __builtin_amdgcn_wmma_f32_16x16x4_f32(false, a, false, bb, (short)0, acc, false, false);
    }
    const int n = n0 + l15;
    const float en = e2[n];
#pragma unroll
    for (int r = 0; r < 8; ++r) {
      const float d = en - 2.f * acc[r];
      if (d < minv[r] || (d == minv[r] && n < mini[r])) { minv[r] = d; mini[r] = n; }
    }
  }

#pragma unroll
  for (int mask = 1; mask <= 8; mask <<= 1) {
#pragma unroll
    for (int r = 0; r < 8; ++r) {
      float ov = __shfl_xor(minv[r], mask, 32);
      int   oi = __shfl_xor(mini[r], mask, 32);
      if (ov < minv[r] || (ov == minv[r] && oi < mini[r])) { minv[r] = ov; mini[r] = oi; }
    }
  }
  if (l15 == 0) {
#pragma unroll
    for (int r = 0; r < 8; ++r) {
      redv[wave * 16 + half * 8 + r] = minv[r];
      redi[wave * 16 + half * 8 + r] = mini[r];
    }
  }
  __syncthreads();
  if (tid < 16) {
    float bv = redv[tid];
    int   bi = redi[tid];
    for (int wv = 1; wv < 8; ++wv) {
      const float ov = redv[wv * 16 + tid];
      const int   oi = redi[wv * 16 + tid];
      if (ov < bv || (ov == bv && oi < bi)) { bv = ov; bi = oi; }
    }
    idx[row0 + tid] = bi;
  }
}

// ---------------------------------------------------------------------------
// Gather quantized vectors into [B,H,T], emit state_indices (as float) and
// per-block partial sums of (q - x)^2 for the commitment loss.
// ---------------------------------------------------------------------------
__global__ __launch_bounds__(256)
void vq_gather_kernel(const float* __restrict__ z, const float* __restrict__ emb,
                      const int* __restrict__ idx, float* __restrict__ zq,
                      float* __restrict__ sind, float* __restrict__ partials) {
  __shared__ float sm[256];
  const size_t e = (size_t)blockIdx.x * 256 + threadIdx.x;
  const int t = (int)(e % Tlen);
  const size_t bh = e / Tlen;
  const int h = (int)(bh % Hdim);
  const int b = (int)(bh / Hdim);
  const int bt = b * Tlen + t;
  const int code = idx[bt];
  const float q = emb[(size_t)code * Hdim + h];
  const float xv = z[e];
  zq[e] = q;
  if (h == 0) sind[bt] = (float)code;
  const float d = q - xv;
  sm[threadIdx.x] = d * d;
  __syncthreads();
  for (int s = 128; s > 0; s >>= 1) {
    if (threadIdx.x < s) sm[threadIdx.x] += sm[threadIdx.x + s];
    __syncthreads();
  }
  if (threadIdx.x == 0) partials[blockIdx.x] = sm[0];
}

// Deterministic final reduction: single block, fixed traversal order.
__global__ __launch_bounds__(256)
void loss_reduce_kernel(const float* __restrict__ partials, int n, float* __restrict__ out) {
  __shared__ float sm[256];
  float s = 0.f;
  for (int i = threadIdx.x; i < n; i += 256) s += partials[i];
  sm[threadIdx.x] = s;
  __syncthreads();
  for (int st = 128; st > 0; st >>= 1) {
    if (threadIdx.x < st) sm[threadIdx.x] += sm[threadIdx.x + st];
    __syncthreads();
  }
  if (threadIdx.x == 0)
    out[0] = 0.25f * sm[0] / (float)((size_t)Bsz * Tlen * Hdim);
}

// ---------------------------------------------------------------------------
extern "C" void kernel_launch(void* const* d_in, const int* in_sizes, int n_in,
                              void* d_out, int out_size, void* d_ws, size_t ws_size,
                              hipStream_t stream) {
  (void)in_sizes; (void)n_in; (void)out_size; (void)ws_size;

  const float* x         = (const float*)d_in[0];
  const float* proj_w    = (const float*)d_in[1];
  const float* proj_b    = (const float*)d_in[2];
  const float* enc_c1_w  = (const float*)d_in[3];
  const float* enc_c1_b  = (const float*)d_in[4];
  const float* enc_bn1_g = (const float*)d_in[5];
  const float* enc_bn1_bt= (const float*)d_in[6];
  const float* enc_bn1_m = (const float*)d_in[7];
  const float* enc_bn1_v = (const float*)d_in[8];
  const float* enc_c2_w  = (const float*)d_in[9];
  const float* enc_c2_b  = (const float*)d_in[10];
  const float* enc_bn2_g = (const float*)d_in[11];
  const float* enc_bn2_bt= (const float*)d_in[12];
  const float* enc_bn2_m = (const float*)d_in[13];
  const float* enc_bn2_v = (const float*)d_in[14];
  const float* embeddings= (const float*)d_in[15];
  const float* pr_c1_w   = (const float*)d_in[16];
  const float* pr_c1_b   = (const float*)d_in[17];
  const float* pr_bn1_g  = (const float*)d_in[18];
  const float* pr_bn1_bt = (const float*)d_in[19];
  const float* pr_bn1_m  = (const float*)d_in[20];
  const float* pr_bn1_v  = (const float*)d_in[21];
  const float* pr_c2_w   = (const float*)d_in[22];
  const float* pr_c2_b   = (const float*)d_in[23];
  const float* pr_bn2_g  = (const float*)d_in[24];
  const float* pr_bn2_bt = (const float*)d_in[25];
  const float* pr_bn2_m  = (const float*)d_in[26];
  const float* pr_bn2_v  = (const float*)d_in[27];
  const float* pr_out_w  = (const float*)d_in[28];
  const float* pr_out_b  = (const float*)d_in[29];

  const size_t nAct = (size_t)Bsz * Hdim * Tlen;   // 33,554,432 floats

  // workspace carve-up (all 4-byte elements)
  float* ws   = (float*)d_ws;
  float* zA   = ws;                 // [B,H,T]
  float* zB   = zA + nAct;          // [B,H,T]
  float* e2   = zB + nAct;          // [K]
  float* eff  = e2 + Kcb;           // 10 layers * (256 scale + 256 shift)
  int*   idx  = (int*)(eff + 10 * 512);
  float* partials = (float*)(idx + (size_t)Bsz * Tlen);   // nAct/256
  float* embT = partials + nAct / 256;                    // [H,K]
  float* wtbuf = embT + (size_t)Hdim * Kcb;               // transposed weights
  float* wt_proj = wtbuf;                                 // 12*256
  float* wt_enc1 = wt_proj + 12 * 256;                    // 4 * 5*256*256
  float* wt_enc2 = wt_enc1 + 4 * 5 * 256 * 256;           // 4 * 5*256*256
  float* wt_pr1  = wt_enc2 + 4 * 5 * 256 * 256;           // 3*256*256
  float* wt_pr2  = wt_pr1 + 3 * 256 * 256;                // 3*256*256
  float* wt_out  = wt_pr2 + 3 * 256 * 256;                // 256*256

  // d_out layout: z_pred | vq_loss | state_indices | z_q
  float* out_pred = (float*)d_out;
  float* out_loss = out_pred + nAct;
  float* out_sind = out_loss + 1;
  float* out_zq   = out_sind + (size_t)Bsz * Tlen;

  // --- parameter preprocessing ---
  for (int i = 0; i < 4; ++i) {
    bn_fold_kernel<<<1, 256, 0, stream>>>(enc_c1_b + i * Hdim, enc_bn1_g + i * Hdim,
                                          enc_bn1_bt + i * Hdim, enc_bn1_m + i * Hdim,
                                          enc_bn1_v + i * Hdim,
                                          eff + (2 * i) * 512, eff + (2 * i) * 512 + 256);
    bn_fold_kernel<<<1, 256, 0, stream>>>(enc_c2_b + i * Hdim, enc_bn2_g + i * Hdim,
                                          enc_bn2_bt + i * Hdim, enc_bn2_m + i * Hdim,
                                          enc_bn2_v + i * Hdim,
                                          eff + (2 * i + 1) * 512, eff + (2 * i + 1) * 512 + 256);
  }
  bn_fold_kernel<<<1, 256, 0, stream>>>(pr_c1_b, pr_bn1_g, pr_bn1_bt, pr_bn1_m, pr_bn1_v,
                                        eff + 8 * 512, eff + 8 * 512 + 256);
  bn_fold_kernel<<<1, 256, 0, stream>>>(pr_c2_b, pr_bn2_g, pr_bn2_bt, pr_bn2_m, pr_bn2_v,
                                        eff + 9 * 512, eff + 9 * 512 + 256);
  code_norm_kernel<<<2, 256, 0, stream>>>(embeddings, e2);
  etrans_kernel<<<(Hdim * Kcb) / 256, 256, 0, stream>>>(embeddings, embT);

  auto wt_launch = [&](const float* w, float* wt, int C, int k) {
    const int total = 256 * C * k;
    wtrans_kernel<<<(total + 255) / 256, 256, 0, stream>>>(w, wt, C, k);
  };
  wt_launch(proj_w, wt_proj, 12, 1);
  for (int i = 0; i < 4; ++i) {
    wt_launch(enc_c1_w + (size_t)i * Hdim * Hdim * 5, wt_enc1 + (size_t)i * 5 * 256 * 256, Hdim, 5);
    wt_launch(enc_c2_w + (size_t)i * Hdim * Hdim * 5, wt_enc2 + (size_t)i * 5 * 256 * 256, Hdim, 5);
  }
  wt_launch(pr_c1_w, wt_pr1, Hdim, 3);
  wt_launch(pr_c2_w, wt_pr2, Hdim, 3);
  wt_launch(pr_out_w, wt_out, Hdim, 1);

  const dim3 cgrid(Tlen / CONV_BLK_N, Hdim / 128, Bsz);  // (128, 2, 16)

  // --- encoder ---
  conv_wmma_kernel<<<cgrid, 256, 0, stream>>>(x, wt_proj, nullptr, proj_b, nullptr,
                                              zA, 12, 1, 1, 0);
  for (int i = 0; i < 4; ++i) {
    const int dil = 1 << i;
    conv_wmma_kernel<<<cgrid, 256, 0, stream>>>(
        zA, wt_enc1 + (size_t)i * 5 * 256 * 256,
        eff + (2 * i) * 512, eff + (2 * i) * 512 + 256, nullptr, zB, Hdim, 5, dil, 1);
    conv_wmma_kernel<<<cgrid, 256, 0, stream>>>(
        zB, wt_enc2 + (size_t)i * 5 * 256 * 256,
        eff + (2 * i + 1) * 512, eff + (2 * i + 1) * 512 + 256, zA, zA, Hdim, 5, dil, 1);
  }

  // --- VQ ---
  vq_argmin_kernel<<<(Bsz * Tlen) / 16, 256, 0, stream>>>(zA, embT, e2, idx);
  const int nblk = (int)(nAct / 256);  // 131072
  vq_gather_kernel<<<nblk, 256, 0, stream>>>(zA, embeddings, idx, out_zq, out_sind, partials);
  loss_reduce_kernel<<<1, 256, 0, stream>>>(partials, nblk, out_loss);

  // --- predictor (reads z_q straight from d_out) ---
  conv_wmma_kernel<<<cgrid, 256, 0, stream>>>(out_zq, wt_pr1,
                                              eff + 8 * 512, eff + 8 * 512 + 256,
                                              nullptr, zB, Hdim, 3, 1, 1);
  conv_wmma_kernel<<<cgrid, 256, 0, stream>>>(zB, wt_pr2,
                                              eff + 9 * 512, eff + 9 * 512 + 256,
                                              out_zq, zA, Hdim, 3, 1, 1);
  conv_wmma_kernel<<<cgrid, 256, 0, stream>>>(zA, wt_out, nullptr, pr_out_b,
                                              nullptr, out_pred, Hdim, 1, 1, 0);
}